// TreeDecoderRNN_712964571854
// MI455X (gfx1250) — compile-verified
//
#include <hip/hip_runtime.h>
#include <hip/hip_bf16.h>
#include <math.h>

#define HDIM 1024
#define VDIM 32000

typedef __attribute__((ext_vector_type(2))) float v2f;
typedef __attribute__((ext_vector_type(8))) float v8f;

__device__ __forceinline__ float sigmoidf_(float x) {
    return 1.0f / (1.0f + __expf(-x));
}

// ---------------------------------------------------------------------------
// One tree level: N parents -> 2N children (interleaved: left=2p, right=2p+1).
// Grid: (ceil(N/16), HDIM/16), block: 32 (one wave).
// Each wave computes a 16(parents) x 16(hidden cols) tile of all six gate
// pre-activations (r,z,n for left & right) via V_WMMA_F32_16X16X4_F32 chains,
// then applies the GRU gating elementwise and scatters both children.
// ---------------------------------------------------------------------------
__global__ __launch_bounds__(32)
void gru_level_kernel(const float* __restrict__ hin,   // N x H parents
                      const float* __restrict__ Wl,    // 3H x H
                      const float* __restrict__ bihl,
                      const float* __restrict__ bhhl,
                      const float* __restrict__ Wr,    // 3H x H
                      const float* __restrict__ bihr,
                      const float* __restrict__ bhhr,
                      float* __restrict__ hout,        // 2N x H children
                      int N)
{
    const int lane = threadIdx.x;        // 0..31 (wave32)
    const int half = lane >> 4;          // 0 or 1
    const int r16  = lane & 15;
    const int mbase = blockIdx.x * 16;   // parent-row tile base
    const int cbase = blockIdx.y * 16;   // hidden-column tile base

    v8f cr_l = {}; v8f cz_l = {}; v8f cn_l = {};
    v8f cr_r = {}; v8f cz_r = {}; v8f cn_r = {};

    const int arow   = mbase + r16;
    const bool rowOK = (arow < N);
    const float* aptr = hin + (size_t)(rowOK ? arow : 0) * HDIM;

    const int col = cbase + r16;
    // B[k][j] = W[j][k]  (gh = h @ W^T), contiguous in k within a W row.
    const float* blr = Wl + (size_t)(0 * HDIM + col) * HDIM;
    const float* blz = Wl + (size_t)(1 * HDIM + col) * HDIM;
    const float* bln = Wl + (size_t)(2 * HDIM + col) * HDIM;
    const float* brr = Wr + (size_t)(0 * HDIM + col) * HDIM;
    const float* brz = Wr + (size_t)(1 * HDIM + col) * HDIM;
    const float* brn = Wr + (size_t)(2 * HDIM + col) * HDIM;

    for (int k = 0; k < HDIM; k += 4) {
        const int ko = k + 2 * half;     // lanes 16-31 carry K=k+2,k+3
        v2f a = *(const v2f*)(aptr + ko);
        if (!rowOK) { a[0] = 0.0f; a[1] = 0.0f; }
        v2f b0 = *(const v2f*)(blr + ko);
        v2f b1 = *(const v2f*)(blz + ko);
        v2f b2 = *(const v2f*)(bln + ko);
        v2f b3 = *(const v2f*)(brr + ko);
        v2f b4 = *(const v2f*)(brz + ko);
        v2f b5 = *(const v2f*)(brn + ko);
        cr_l = __builtin_amdgcn_wmma_f32_16x16x4_f32(false, a, false, b0, (short)0, cr_l, false, false);
        cz_l = __builtin_amdgcn_wmma_f32_16x16x4_f32(false, a, false, b1, (short)0, cz_l, false, false);
        cn_l = __builtin_amdgcn_wmma_f32_16x16x4_f32(false, a, false, b2, (short)0, cn_l, false, false);
        cr_r = __builtin_amdgcn_wmma_f32_16x16x4_f32(false, a, false, b3, (short)0, cr_r, false, false);
        cz_r = __builtin_amdgcn_wmma_f32_16x16x4_f32(false, a, false, b4, (short)0, cz_r, false, false);
        cn_r = __builtin_amdgcn_wmma_f32_16x16x4_f32(false, a, false, b5, (short)0, cn_r, false, false);
    }

    const float bihl_r = bihl[col], bihl_z = bihl[HDIM + col], bihl_n = bihl[2 * HDIM + col];
    const float bhhl_r = bhhl[col], bhhl_z = bhhl[HDIM + col], bhhl_n = bhhl[2 * HDIM + col];
    const float bihr_r = bihr[col], bihr_z = bihr[HDIM + col], bihr_n = bihr[2 * HDIM + col];
    const float bhhr_r = bhhr[col], bhhr_z = bhhr[HDIM + col], bhhr_n = bhhr[2 * HDIM + col];

    #pragma unroll
    for (int i = 0; i < 8; ++i) {
        const int p = mbase + i + 8 * half;      // global parent row for slot i
        if (p >= N) continue;
        const float hp = hin[(size_t)p * HDIM + col];
        // left child
        {
            const float ghr = cr_l[i] + bhhl_r;
            const float r   = sigmoidf_(bihl_r + ghr);
            const float ghz = cz_l[i] + bhhl_z;
            const float z   = sigmoidf_(bihl_z + ghz);
            const float ghn = cn_l[i] + bhhl_n;
            const float n   = tanhf(bihl_n + r * ghn);
            hout[(size_t)(2 * p) * HDIM + col] = (1.0f - z) * n + z * hp;
        }
        // right child
        {
            const float ghr = cr_r[i] + bhhr_r;
            const float r   = sigmoidf_(bihr_r + ghr);
            const float ghz = cz_r[i] + bhhr_z;
            const float z   = sigmoidf_(bihr_z + ghz);
            const float ghn = cn_r[i] + bhhr_n;
            const float n   = tanhf(bihr_n + r * ghn);
            hout[(size_t)(2 * p + 1) * HDIM + col] = (1.0f - z) * n + z * hp;
        }
    }
}

// ---------------------------------------------------------------------------
// logits = h(1024 x H) @ Wout^T (H x V) + bout.  One wave computes a 64x32
// block (4x2 tiles of 16x16) -> Wout rows re-read only 16x (L2-resident).
// Grid: (1024/64, VDIM/32), block: 32.
// ---------------------------------------------------------------------------
#define TM 4
#define TN 2
__global__ __launch_bounds__(32)
void out_gemm_kernel(const float* __restrict__ hA,    // 1024 x H
                     const float* __restrict__ Wout,  // V x H
                     const float* __restrict__ bout,  // V
                     float* __restrict__ out)         // 1024 x V
{
    const int lane = threadIdx.x;
    const int half = lane >> 4;
    const int r16  = lane & 15;
    const int mbase = blockIdx.x * (16 * TM);
    const int nbase = blockIdx.y * (16 * TN);

    v8f acc[TM][TN];
    #pragma unroll
    for (int tm = 0; tm < TM; ++tm)
        #pragma unroll
        for (int tn = 0; tn < TN; ++tn)
            acc[tm][tn] = (v8f){};

    const float* ap[TM];
    const float* bp[TN];
    #pragma unroll
    for (int tm = 0; tm < TM; ++tm)
        ap[tm] = hA + (size_t)(mbase + tm * 16 + r16) * HDIM;
    #pragma unroll
    for (int tn = 0; tn < TN; ++tn)
        bp[tn] = Wout + (size_t)(nbase + tn * 16 + r16) * HDIM;

    for (int k = 0; k < HDIM; k += 4) {
        const int ko = k + 2 * half;
        v2f a[TM], b[TN];
        #pragma unroll
        for (int tm = 0; tm < TM; ++tm) a[tm] = *(const v2f*)(ap[tm] + ko);
        #pragma unroll
        for (int tn = 0; tn < TN; ++tn) b[tn] = *(const v2f*)(bp[tn] + ko);
        #pragma unroll
        for (int tm = 0; tm < TM; ++tm)
            #pragma unroll
            for (int tn = 0; tn < TN; ++tn)
                acc[tm][tn] = __builtin_amdgcn_wmma_f32_16x16x4_f32(
                    false, a[tm], false, b[tn], (short)0, acc[tm][tn], false, false);
    }

    #pragma unroll
    for (int tn = 0; tn < TN; ++tn) {
        const int col = nbase + tn * 16 + r16;
        const float bb = bout[col];
        #pragma unroll
        for (int tm = 0; tm < TM; ++tm) {
            #pragma unroll
            for (int i = 0; i < 8; ++i) {
                const int row = mbase + tm * 16 + i + 8 * half;
                out[(size_t)row * VDIM + col] = acc[tm][tn][i] + bb;
            }
        }
    }
}

// ---------------------------------------------------------------------------
// In-place row-wise log-softmax over V=32000.  One block per row; online
// (max, sumexp) per thread, LDS tree reduction, second pass subtracts LSE.
// Row (128 KB) is L2-resident between the two passes.
// ---------------------------------------------------------------------------
__global__ __launch_bounds__(256)
void logsoftmax_kernel(float* __restrict__ out)
{
    const int row = blockIdx.x;
    float* x = out + (size_t)row * VDIM;

    float m = -INFINITY, s = 0.0f;
    for (int j = threadIdx.x; j < VDIM; j += 256) {
        const float v  = x[j];
        const float mn = fmaxf(m, v);
        s = s * __expf(m - mn) + __expf(v - mn);
        m = mn;
    }

    __shared__ float sm[256];
    __shared__ float ss[256];
    sm[threadIdx.x] = m;
    ss[threadIdx.x] = s;
    __syncthreads();
    for (int off = 128; off > 0; off >>= 1) {
        if (threadIdx.x < off) {
            const float m1 = sm[threadIdx.x], s1 = ss[threadIdx.x];
            const float m2 = sm[threadIdx.x + off], s2 = ss[threadIdx.x + off];
            const float mn = fmaxf(m1, m2);
            sm[threadIdx.x] = mn;
            ss[threadIdx.x] = s1 * __expf(m1 - mn) + s2 * __expf(m2 - mn);
        }
        __syncthreads();
    }
    const float lse = sm[0] + __logf(ss[0]);
    __syncthreads();

    for (int j = threadIdx.x; j < VDIM; j += 256)
        x[j] -= lse;
}

// ---------------------------------------------------------------------------
extern "C" void kernel_launch(void* const* d_in, const int* in_sizes, int n_in,
                              void* d_out, int out_size, void* d_ws, size_t ws_size,
                              hipStream_t stream) {
    (void)in_sizes; (void)n_in; (void)out_size; (void)ws_size;
    const float* hidden = (const float*)d_in[0];
    const float* Wl     = (const float*)d_in[1];
    const float* bihl   = (const float*)d_in[2];
    const float* bhhl   = (const float*)d_in[3];
    const float* Wr     = (const float*)d_in[4];
    const float* bihr   = (const float*)d_in[5];
    const float* bhhr   = (const float*)d_in[6];
    const float* Wout   = (const float*)d_in[7];
    const float* bout   = (const float*)d_in[8];
    float* out = (float*)d_out;

    // Ping-pong node-hidden buffers in workspace: 2 x (1024 x 1024) f32 = 8 MB.
    float* buf0 = (float*)d_ws;
    float* buf1 = buf0 + (size_t)1024 * HDIM;

    // Root hidden state -> row 0 of buf0.
    hipMemcpyAsync(buf0, hidden, HDIM * sizeof(float),
                   hipMemcpyDeviceToDevice, stream);

    float* cur = buf0;
    float* nxt = buf1;
    int N = 1;
    for (int d = 0; d < 10; ++d) {           // DEPTH = 10 (reference constant)
        dim3 grid((N + 15) / 16, HDIM / 16);
        gru_level_kernel<<<grid, 32, 0, stream>>>(cur, Wl, bihl, bhhl,
                                                  Wr, bihr, bhhr, nxt, N);
        float* t = cur; cur = nxt; nxt = t;
        N *= 2;
    }
    // cur now holds the 1024 x H leaf hidden states.

    dim3 g2(1024 / (16 * TM), VDIM / (16 * TN));
    out_gemm_kernel<<<g2, 32, 0, stream>>>(cur, Wout, bout, out);

    logsoftmax_kernel<<<1024, 256, 0, stream>>>(out);
}